// PalletLoss_3564822856246
// MI455X (gfx1250) — compile-verified
//
#include <hip/hip_runtime.h>
#include <hip/hip_bf16.h>
#include <math.h>

// ---------------------------------------------------------------------------
// PalletLoss on MI455X (gfx1250, wave32, WMMA)
//   pallet: (1024, 8, 3) f32,  comp: (1024, 8, 3) f32,  out: scalar f32
//   N = M = 8192 colors. Distance tiles via V_WMMA_F32_16X16X4_F32:
//     A row m  = (-2*bx, -2*by, -2*bz, 1)        (keys, from comp)
//     B col n  = ( ax,    ay,    az,   a2 )      (queries, from sorted pallet)
//     C[m][n]  =  b2[m]
//     D = sq distances = b2 + a2 - 2*<b,a>
//   LDS holds comp as AoS float4 (x, y, z, |b|^2): one ds_load_b128 per key
//   row feeds both the WMMA C seed (.w) and the weighted-sum FMAs (.xyz).
//   Epilogue uses raw HW transcendentals (v_sqrt_f32 / v_exp_f32).
// ---------------------------------------------------------------------------

typedef __attribute__((ext_vector_type(2))) float v2f;
typedef __attribute__((ext_vector_type(8))) float v8f;

#define S_DIM   1024
#define P_DIM   8
#define NCOL    8192          // S_DIM * P_DIM
#define NTILES  512           // NCOL / 16
#define WAVES   8             // waves per block in main kernel

#define NEG_LOG2E (-1.4426950408889634f)

// ---------------------------------------------------------------------------
// Kernel 0: per-column stable sort of pallet by luma norm (rank by counting).
// grid = 8 (one block per pallet column), block = 1024 threads.
// ---------------------------------------------------------------------------
__global__ void pallet_sort_kernel(const float* __restrict__ pallet,
                                   float* __restrict__ a_sorted) {
    __shared__ float keys[S_DIM];
    const int p = blockIdx.x;       // pallet column 0..7
    const int s = threadIdx.x;      // row 0..1023

    const float inv = 0.5f;         // 1 / PALLET_INERTIA
    float cx = fminf(fmaxf(pallet[(s * P_DIM + p) * 3 + 0] * inv, 0.f), 1.f);
    float cy = fminf(fmaxf(pallet[(s * P_DIM + p) * 3 + 1] * inv, 0.f), 1.f);
    float cz = fminf(fmaxf(pallet[(s * P_DIM + p) * 3 + 2] * inv, 0.f), 1.f);
    float key = cx * cx * 0.299f + cy * cy * 0.587f + cz * cz * 0.114f;

    keys[s] = key;
    __syncthreads();

    // stable ascending rank
    int rank = 0;
    for (int j = 0; j < S_DIM; ++j) {
        float kj = keys[j];
        rank += (kj < key) || (kj == key && j < s);
    }

    float* dst = a_sorted + (rank * P_DIM + p) * 3;
    dst[0] = cx; dst[1] = cy; dst[2] = cz;
}

// ---------------------------------------------------------------------------
// Kernel 1: softmin-weighted nearest color + per-block MSE partial.
// grid = 512 blocks (one per 16-query tile), block = 256 threads (8 waves).
// Dynamic LDS: comp as float4 AoS (8192 * 16B = 128KB) + wave partials.
// ---------------------------------------------------------------------------
__global__ void pallet_dist_kernel(const float* __restrict__ comp,
                                   const float* __restrict__ a_sorted,
                                   float* __restrict__ partials) {
    extern __shared__ float smem[];
    float4* bv   = (float4*)smem;              // [8192] (x, y, z, |b|^2)
    float* wpart = smem + 4 * NCOL;            // [WAVES * 16 * 4]

    const int tid  = threadIdx.x;
    const int wave = tid >> 5;
    const int lane = tid & 31;
    const int half = lane >> 4;       // 0: K=0,1 / rows m..m+7 ; 1: K=2,3 / rows m+8..
    const int col  = lane & 15;

    // ---- stage comp into LDS (AoS float4 with precomputed squared norm) ---
    for (int m = tid; m < NCOL; m += blockDim.x) {
        float x = comp[m * 3 + 0];
        float y = comp[m * 3 + 1];
        float z = comp[m * 3 + 2];
        bv[m] = make_float4(x, y, z, x * x + y * y + z * z);
    }
    __syncthreads();

    // ---- B operand (queries) is fixed for the whole block -----------------
    const int n = blockIdx.x * 16 + col;
    const float qx = a_sorted[n * 3 + 0];
    const float qy = a_sorted[n * 3 + 1];
    const float qz = a_sorted[n * 3 + 2];
    const float q2 = qx * qx + qy * qy + qz * qz;
    v2f bmat;
    bmat[0] = half ? qz : qx;         // K=2 : K=0
    bmat[1] = half ? q2 : qy;         // K=3 : K=1

    float denom = 0.f, nx = 0.f, ny = 0.f, nz = 0.f;

    // ---- sweep all 8192 keys in 16-row tiles, 64 tiles per wave -----------
    for (int t = wave; t < NTILES; t += WAVES) {
        const int base = t * 16;

        // A operand: one b128 load per lane (row m = base + col)
        const float4 ba = bv[base + col];
        v2f amat;
        amat[0] = half ? (-2.0f * ba.z) : (-2.0f * ba.x);   // K=2 : K=0
        amat[1] = half ? 1.0f           : (-2.0f * ba.y);   // K=3 : K=1

        // 8 broadcast b128 loads cover C seed (.w) and epilogue (.xyz)
        const int rbase = base + half * 8;
        float4 bm[8];
#pragma unroll
        for (int v = 0; v < 8; ++v) bm[v] = bv[rbase + v];

        v8f c;
#pragma unroll
        for (int v = 0; v < 8; ++v) c[v] = bm[v].w;

        // D = b2 + a2 - 2<b,a>  (16x16 tile of squared distances)
        v8f d = __builtin_amdgcn_wmma_f32_16x16x4_f32(
            false, amat, false, bmat, (short)0, c, false, false);

#pragma unroll
        for (int v = 0; v < 8; ++v) {
            // clamp(d, 0, +big) in one v_med3_f32; raw v_sqrt_f32 + v_exp_f32
            float sq = __builtin_amdgcn_fmed3f(d[v], 0.0f, 3.0e38f);
            float t1 = __builtin_amdgcn_sqrtf(sq);
            float e  = __builtin_amdgcn_exp2f(t1 * NEG_LOG2E);   // exp(-dist)
            denom += e;
            nx = fmaf(e, bm[v].x, nx);
            ny = fmaf(e, bm[v].y, ny);
            nz = fmaf(e, bm[v].z, nz);
        }
    }

    // ---- fold the two lane-halves (same n, disjoint m rows) ---------------
    denom += __shfl_xor(denom, 16, 32);
    nx    += __shfl_xor(nx, 16, 32);
    ny    += __shfl_xor(ny, 16, 32);
    nz    += __shfl_xor(nz, 16, 32);

    if (lane < 16) {
        float* wp = wpart + (wave * 16 + col) * 4;
        wp[0] = denom; wp[1] = nx; wp[2] = ny; wp[3] = nz;
    }
    __syncthreads();

    // ---- cross-wave fold + per-query MSE, block partial -------------------
    if (tid < 16) {
        float dsum = 0.f, sx = 0.f, sy = 0.f, sz = 0.f;
#pragma unroll
        for (int w = 0; w < WAVES; ++w) {
            const float* wp = wpart + (w * 16 + tid) * 4;
            dsum += wp[0]; sx += wp[1]; sy += wp[2]; sz += wp[3];
        }
        const float r = __builtin_amdgcn_rcpf(dsum);
        const int nn = blockIdx.x * 16 + tid;
        float ex = a_sorted[nn * 3 + 0] - sx * r;
        float ey = a_sorted[nn * 3 + 1] - sy * r;
        float ez = a_sorted[nn * 3 + 2] - sz * r;
        float mse = ex * ex + ey * ey + ez * ez;

        // reduce 16 lanes -> lane 0 (still inside wave 0)
        mse += __shfl_down(mse, 8, 32);
        mse += __shfl_down(mse, 4, 32);
        mse += __shfl_down(mse, 2, 32);
        mse += __shfl_down(mse, 1, 32);
        if (tid == 0) partials[blockIdx.x] = mse;
    }
}

// ---------------------------------------------------------------------------
// Kernel 2: deterministic tree reduction of 512 block partials -> mean.
// ---------------------------------------------------------------------------
__global__ void pallet_reduce_kernel(const float* __restrict__ partials,
                                     float* __restrict__ out) {
    __shared__ float red[256];
    const int tid = threadIdx.x;
    float s = partials[tid] + partials[tid + 256];
    red[tid] = s;
    __syncthreads();
    for (int off = 128; off > 0; off >>= 1) {
        if (tid < off) red[tid] += red[tid + off];
        __syncthreads();
    }
    if (tid == 0) out[0] = red[0] * (1.0f / (float)(NCOL * 3));
}

// ---------------------------------------------------------------------------
extern "C" void kernel_launch(void* const* d_in, const int* in_sizes, int n_in,
                              void* d_out, int out_size, void* d_ws, size_t ws_size,
                              hipStream_t stream) {
    const float* pallet = (const float*)d_in[0];   // (1024, 8, 3)
    const float* comp   = (const float*)d_in[1];   // (1024, 8, 3)
    float* out = (float*)d_out;

    float* ws       = (float*)d_ws;
    float* a_sorted = ws;                 // 8192 * 3 floats
    float* partials = ws + NCOL * 3;      // 512 floats

    pallet_sort_kernel<<<P_DIM, S_DIM, 0, stream>>>(pallet, a_sorted);

    const size_t smem = (size_t)(4 * NCOL + WAVES * 16 * 4) * sizeof(float);
    pallet_dist_kernel<<<NTILES, 32 * WAVES, smem, stream>>>(comp, a_sorted, partials);

    pallet_reduce_kernel<<<1, 256, 0, stream>>>(partials, out);
}